// Group_41979010351621
// MI455X (gfx1250) — compile-verified
//
#include <hip/hip_runtime.h>
#include <hip/hip_bf16.h>
#include <stddef.h>
#include <stdint.h>

// ---------------- problem constants ----------------
#define BATCH   32
#define NPTS    8192
#define NGRP    512
#define KNN_K   32
#define MACRO   128                 // points per TDM tile
#define NT_MACRO (NPTS / MACRO)     // 64 tiles

typedef __attribute__((ext_vector_type(2))) float v2f;
typedef __attribute__((ext_vector_type(8))) float v8f;
typedef __attribute__((ext_vector_type(4))) unsigned int u32x4;
typedef __attribute__((ext_vector_type(8))) int          i32x8;
typedef __attribute__((ext_vector_type(4))) int          i32x4;

// =====================================================================
// TDM: 1D copy of one 2KB tile (512 x 4B elements) global -> LDS.
// D# per CDNA5 ISA ch.8: group0 {count=1, lds_addr, global_addr, type=2},
// group1 {data_size=4B, tensor_dim0=512, tensor_dim1=1, tile_dim0=512,
//         tensor_dim0_stride=512}.
// =====================================================================
__device__ __forceinline__ void tdm_load_tile(unsigned long long gaddr,
                                              unsigned int lds_addr) {
    u32x4 g0;
    g0[0] = 1u;                                       // count=1 (valid user D#)
    g0[1] = lds_addr;                                 // LDS byte address
    g0[2] = (unsigned int)gaddr;                      // global_addr[31:0]
    g0[3] = (unsigned int)(gaddr >> 32) | (2u << 30); // global_addr[56:32] | type=2
    i32x8 g1;
    g1[0] = 0x00020000;             // workgroup_mask=0, data_size=2 (4 bytes)
    g1[1] = (int)(512u << 16);      // tensor_dim0[15:0] = 512
    g1[2] = 0x00010000;             // tensor_dim0 hi = 0, tensor_dim1 = 1
    g1[3] = (int)(512u << 16);      // tensor_dim1 hi = 0, tile_dim0 = 512
    g1[4] = 0;                      // tile_dim1 = 0 (unused), tile_dim2 = 0
    g1[5] = 512;                    // tensor_dim0_stride = 512
    g1[6] = 0;
    g1[7] = 0;
    i32x4 g2 = {0, 0, 0, 0};
    i32x4 g3 = {0, 0, 0, 0};
#if defined(__clang_major__) && __clang_major__ >= 23
    i32x8 gz = {0, 0, 0, 0, 0, 0, 0, 0};
    __builtin_amdgcn_tensor_load_to_lds(g0, g1, g2, g3, gz, 0);
#else
    __builtin_amdgcn_tensor_load_to_lds(g0, g1, g2, g3, 0);
#endif
}

// =====================================================================
// Kernel 0: pack points as float4 (x, y, z, |p|^2). Computed once,
// consumed 8x by the KNN blocks (whole array lives in L2).
// =====================================================================
__global__ __launch_bounds__(256) void pack_kernel(const float* __restrict__ xyz,
                                                   float4* __restrict__ packed) {
    const size_t t = (size_t)blockIdx.x * blockDim.x + threadIdx.x;
    if (t >= (size_t)BATCH * NPTS) return;
    const float x = xyz[t * 3 + 0];
    const float y = xyz[t * 3 + 1];
    const float z = xyz[t * 3 + 2];
    float4 o;
    o.x = x; o.y = y; o.z = z; o.w = x * x + y * y + z * z;
    packed[t] = o;
}

// =====================================================================
// Kernel 1: Farthest Point Sampling. One 1024-thread block per batch.
// Each thread keeps its 8 points + running min-distance in registers.
// =====================================================================
__global__ __launch_bounds__(1024) void fps_kernel(const float* __restrict__ xyz,
                                                   float* __restrict__ centers) {
    const int b   = blockIdx.x;
    const int tid = threadIdx.x;
    const float* base = xyz + (size_t)b * NPTS * 3;

    float px[8], py[8], pz[8], dmin[8];
#pragma unroll
    for (int i = 0; i < 8; ++i) {
        int n = tid * 8 + i;
        px[i] = base[n * 3 + 0];
        py[i] = base[n * 3 + 1];
        pz[i] = base[n * 3 + 2];
        dmin[i] = 1e10f;
    }

    __shared__ float s_cx, s_cy, s_cz;
    __shared__ int   s_far;
    __shared__ float s_rv[32];
    __shared__ int   s_ri[32];

    if (tid == 0) s_far = 0;
    __syncthreads();

    for (int it = 0; it < NGRP; ++it) {
        const int far = s_far;
        if ((far >> 3) == tid) {                 // owner broadcasts centroid
            const int e = far & 7;
            float cx = 0.f, cy = 0.f, cz = 0.f;
#pragma unroll
            for (int i = 0; i < 8; ++i)
                if (e == i) { cx = px[i]; cy = py[i]; cz = pz[i]; }
            s_cx = cx; s_cy = cy; s_cz = cz;
            float* co = centers + ((size_t)b * NGRP + it) * 3;
            co[0] = cx; co[1] = cy; co[2] = cz;
        }
        __syncthreads();
        const float cx = s_cx, cy = s_cy, cz = s_cz;

        float bv = -1.f; int bi = 0;
#pragma unroll
        for (int i = 0; i < 8; ++i) {
            float dx = px[i] - cx, dy = py[i] - cy, dz = pz[i] - cz;
            float d  = dx * dx + dy * dy + dz * dz;
            dmin[i]  = fminf(dmin[i], d);
            int n = tid * 8 + i;
            if (dmin[i] > bv || (dmin[i] == bv && n < bi)) { bv = dmin[i]; bi = n; }
        }
#pragma unroll
        for (int off = 16; off > 0; off >>= 1) {   // wave32 argmax
            float ov = __shfl_xor(bv, off, 32);
            int   oi = __shfl_xor(bi, off, 32);
            if (ov > bv || (ov == bv && oi < bi)) { bv = ov; bi = oi; }
        }
        const int wv = tid >> 5;
        if ((tid & 31) == 0) { s_rv[wv] = bv; s_ri[wv] = bi; }
        __syncthreads();
        if (tid < 32) {
            float v = s_rv[tid];
            int   i = s_ri[tid];
#pragma unroll
            for (int off = 16; off > 0; off >>= 1) {
                float ov = __shfl_xor(v, off, 32);
                int   oi = __shfl_xor(i, off, 32);
                if (ov > v || (ov == v && oi < i)) { v = ov; i = oi; }
            }
            if (tid == 0) s_far = i;
        }
        __syncthreads();
    }
}

// =====================================================================
// Kernel 2: KNN. Block = 4 waves; each wave owns a 16-query tile.
// Wave 0 streams 128-point tiles of the packed float4 array into LDS
// with the Tensor Data Mover (double-buffered, s_wait_tensorcnt);
// every wave runs 8 x V_WMMA_F32_16X16X4_F32 per tile, then filtered
// top-32 insertion.  A[m]=(-2cx,-2cy,-2cz,1), Bcol[n]=(x,y,z,|p|^2)
// => D = |p|^2 - 2 c.p (row-wise ordering identical to true d^2).
// =====================================================================
__global__ __launch_bounds__(128) void knn_kernel(const float* __restrict__ centers,
                                                  const float4* __restrict__ packed,
                                                  int* __restrict__ knn_idx) {
    const int b    = blockIdx.x;
    const int wid  = threadIdx.x >> 5;
    const int lane = threadIdx.x & 31;
    const int mtile = blockIdx.y * 4 + wid;          // 0..31

    __shared__ float4 s_pts[2][MACRO];               // TDM staging (block-shared)
    __shared__ float  s_tile[4][16][MACRO];          // per-wave score tiles
    __shared__ float  s_dist[4][16][KNN_K];
    __shared__ int    s_idx [4][16][KNN_K];

    for (int j = lane; j < 16 * KNN_K; j += 32) {
        s_dist[wid][j >> 5][j & 31] = 3.0e38f;
        s_idx [wid][j >> 5][j & 31] = 0;
    }

    // ---- A fragment (16x4 f32): lanes 0-15 K=0..1, lanes 16-31 K=2..3
    const int m = mtile * 16 + (lane & 15);
    const float* cp = centers + ((size_t)b * NGRP + m) * 3;
    const float cx = cp[0], cy = cp[1], cz = cp[2];
    v2f afrag;
    if (lane < 16) { afrag.x = -2.f * cx; afrag.y = -2.f * cy; }
    else           { afrag.x = -2.f * cz; afrag.y = 1.f; }

    const unsigned long long gbase =
        (unsigned long long)(uintptr_t)(packed + (size_t)b * NPTS);
    const unsigned int lds0 = (unsigned int)(uintptr_t)&s_pts[0][0];
    const unsigned int lds1 = (unsigned int)(uintptr_t)&s_pts[1][0];

    if (wid == 0) tdm_load_tile(gbase, lds0);        // prime the pipeline
    __syncthreads();

    for (int t = 0; t < NT_MACRO; ++t) {
        if (wid == 0) {
            if (t + 1 < NT_MACRO) {
                tdm_load_tile(gbase + (unsigned long long)(t + 1) * MACRO * 16,
                              ((t + 1) & 1) ? lds1 : lds0);
                __builtin_amdgcn_s_wait_tensorcnt(1);  // tile t complete
            } else {
                __builtin_amdgcn_s_wait_tensorcnt(0);
            }
        }
        __syncthreads();                               // buf[t&1] ready for all

        const float4* buf = s_pts[t & 1];
        const int row0 = (lane < 16) ? 0 : 8;
        const int col  = lane & 15;
#pragma unroll
        for (int s = 0; s < 8; ++s) {
            const float4 p = buf[s * 16 + (lane & 15)];   // ds_load per lane
            v2f bfrag;
            if (lane < 16) { bfrag.x = p.x; bfrag.y = p.y; }
            else           { bfrag.x = p.z; bfrag.y = p.w; }
            v8f acc = {};
            acc = __builtin_amdgcn_wmma_f32_16x16x4_f32(
                false, afrag, false, bfrag, (short)0, acc, false, false);
#pragma unroll
            for (int v = 0; v < 8; ++v)
                s_tile[wid][row0 + v][s * 16 + col] = acc[v];
        }
        __syncthreads();       // s_tile complete; also fences buf reuse

        if (lane < 16) {       // filtered top-32 insertion, one row per lane
            float* ld = s_dist[wid][lane];
            int*   li = s_idx [wid][lane];
            float  thr = ld[KNN_K - 1];
#pragma unroll 1
            for (int c = 0; c < MACRO; ++c) {
                const float d = s_tile[wid][lane][c];
                if (d < thr) {
                    int j = KNN_K - 1;
#pragma unroll 1
                    while (j > 0 && ld[j - 1] > d) {
                        ld[j] = ld[j - 1]; li[j] = li[j - 1]; --j;
                    }
                    ld[j] = d; li[j] = t * MACRO + c;
                    thr = ld[KNN_K - 1];
                }
            }
        }
        __syncthreads();
    }

    if (lane < 16) {
        int* out = knn_idx + ((size_t)b * NGRP + mtile * 16 + lane) * KNN_K;
#pragma unroll 1
        for (int j = 0; j < KNN_K; ++j) out[j] = s_idx[wid][lane][j];
    }
}

// =====================================================================
// Kernel 3: greedy nearest-neighbor path over the 512 centers.
// =====================================================================
__global__ __launch_bounds__(512) void order_kernel(const float* __restrict__ centers,
                                                    int* __restrict__ order) {
    const int b = blockIdx.x;
    const int tid = threadIdx.x;

    __shared__ float scx[NGRP], scy[NGRP], scz[NGRP];
    __shared__ unsigned char svis[NGRP];
    __shared__ int   s_last;
    __shared__ float s_rv[16];
    __shared__ int   s_ri[16];

    const float* cb = centers + (size_t)b * NGRP * 3;
    scx[tid] = cb[tid * 3 + 0];
    scy[tid] = cb[tid * 3 + 1];
    scz[tid] = cb[tid * 3 + 2];
    svis[tid] = (tid == 0) ? 1 : 0;
    if (tid == 0) { s_last = 0; order[(size_t)b * NGRP] = 0; }
    __syncthreads();

    const float mx = scx[tid], my = scy[tid], mz = scz[tid];

    for (int it = 1; it < NGRP; ++it) {
        const int last = s_last;
        float dx = mx - scx[last], dy = my - scy[last], dz = mz - scz[last];
        float d = dx * dx + dy * dy + dz * dz;
        if (svis[tid]) d = 3.0e38f;

        float bv = d; int bi = tid;
#pragma unroll
        for (int off = 16; off > 0; off >>= 1) {
            float ov = __shfl_xor(bv, off, 32);
            int   oi = __shfl_xor(bi, off, 32);
            if (ov < bv || (ov == bv && oi < bi)) { bv = ov; bi = oi; }
        }
        if ((tid & 31) == 0) { s_rv[tid >> 5] = bv; s_ri[tid >> 5] = bi; }
        __syncthreads();
        if (tid < 32) {
            float v = (tid < 16) ? s_rv[tid] : 3.0e38f;
            int   i = (tid < 16) ? s_ri[tid] : 0x7fffffff;
#pragma unroll
            for (int off = 16; off > 0; off >>= 1) {
                float ov = __shfl_xor(v, off, 32);
                int   oi = __shfl_xor(i, off, 32);
                if (ov < v || (ov == v && oi < i)) { v = ov; i = oi; }
            }
            if (tid == 0) {
                s_last = i;
                svis[i] = 1;
                order[(size_t)b * NGRP + it] = i;
            }
        }
        __syncthreads();
    }
}

// =====================================================================
// Kernel 4: final gather -> centered neighborhood + reordered centers
// =====================================================================
__global__ __launch_bounds__(256) void gather_kernel(const float* __restrict__ xyz,
                                                     const float* __restrict__ centers,
                                                     const int* __restrict__ knn_idx,
                                                     const int* __restrict__ order,
                                                     float* __restrict__ out_nbh,
                                                     float* __restrict__ out_ctr) {
    const size_t t = (size_t)blockIdx.x * blockDim.x + threadIdx.x;
    const size_t total = (size_t)BATCH * NGRP * KNN_K;
    if (t >= total) return;
    const int j   = (int)(t % KNN_K);
    const int g_i = (int)((t / KNN_K) % NGRP);
    const int b   = (int)(t / ((size_t)KNN_K * NGRP));

    const int g = order[(size_t)b * NGRP + g_i];
    const float* c = centers + ((size_t)b * NGRP + g) * 3;
    const int n = knn_idx[((size_t)b * NGRP + g) * KNN_K + j];
    const float* p = xyz + ((size_t)b * NPTS + n) * 3;

    float* o = out_nbh + t * 3;
    o[0] = p[0] - c[0];
    o[1] = p[1] - c[1];
    o[2] = p[2] - c[2];

    if (j == 0) {
        float* oc = out_ctr + ((size_t)b * NGRP + g_i) * 3;
        oc[0] = c[0]; oc[1] = c[1]; oc[2] = c[2];
    }
}

// =====================================================================
extern "C" void kernel_launch(void* const* d_in, const int* in_sizes, int n_in,
                              void* d_out, int out_size, void* d_ws, size_t ws_size,
                              hipStream_t stream) {
    const float* xyz = (const float*)d_in[0];

    // scratch layout (all 16B-aligned)
    const size_t centers_bytes = (size_t)BATCH * NGRP * 3 * sizeof(float);   // 192 KB
    const size_t knn_bytes     = (size_t)BATCH * NGRP * KNN_K * sizeof(int); // 2 MB
    const size_t order_bytes   = (size_t)BATCH * NGRP * sizeof(int);         // 64 KB
    float*  centers = (float*)d_ws;
    int*    knn_idx = (int*)((char*)d_ws + centers_bytes);
    int*    order   = (int*)((char*)d_ws + centers_bytes + knn_bytes);
    float4* packed  = (float4*)((char*)d_ws + centers_bytes + knn_bytes + order_bytes);

    float* out_nbh = (float*)d_out;
    float* out_ctr = out_nbh + (size_t)BATCH * NGRP * KNN_K * 3;

    pack_kernel<<<(BATCH * NPTS + 255) / 256, 256, 0, stream>>>(xyz, packed);

    fps_kernel<<<BATCH, 1024, 0, stream>>>(xyz, centers);

    dim3 kgrid(BATCH, 8);                       // 8 blocks * 4 waves = 32 query tiles
    knn_kernel<<<kgrid, 128, 0, stream>>>(centers, packed, knn_idx);

    order_kernel<<<BATCH, NGRP, 0, stream>>>(centers, order);

    const size_t total = (size_t)BATCH * NGRP * KNN_K;
    gather_kernel<<<(unsigned)((total + 255) / 256), 256, 0, stream>>>(
        xyz, centers, knn_idx, order, out_nbh, out_ctr);
}